// CurvatureGraphNN_8186207667012
// MI455X (gfx1250) — compile-verified
//
#include <hip/hip_runtime.h>
#include <cmath>
#include <stdint.h>

typedef __attribute__((ext_vector_type(2))) float v2f;
typedef __attribute__((ext_vector_type(4))) float v4f;
typedef __attribute__((ext_vector_type(8))) float v8f;

#define N_NODES 50000
#define N_EDGES 800000
#define D_IN    512
#define D_HID   64
#define D_OUT   40

// ---------------------------------------------------------------------------
// Async global -> LDS copy (CDNA5 GLOBAL_LOAD_ASYNC_TO_LDS_B128, ASYNCcnt).
// lds_off: byte offset into LDS (low 32 bits of a generic shared pointer).
// ---------------------------------------------------------------------------
__device__ __forceinline__ void async_copy_b128(uint32_t lds_off, const void* gaddr) {
    asm volatile("global_load_async_to_lds_b128 %0, %1, off"
                 :: "v"(lds_off), "v"((unsigned long long)(uintptr_t)gaddr)
                 : "memory");
}
__device__ __forceinline__ void wait_async_le4() {
    asm volatile("s_wait_asynccnt 0x4" ::: "memory");
}
__device__ __forceinline__ void wait_async_0() {
    asm volatile("s_wait_asynccnt 0x0" ::: "memory");
}

// ---------------------------------------------------------------------------
// GEMM1: h1[N,64] = x[N,512] @ W1[512,64] + b1   (fp32 WMMA 16x16x4)
// Block = 256 threads = 8 waves; wave computes 16 rows x 64 cols (4 C tiles).
// x is streamed through double-buffered LDS (2 x 16 KB) with async-to-LDS
// copies: chunk c+1 in flight while chunk c feeds the WMMA pipeline.
// ---------------------------------------------------------------------------
#define KC 32                       // K-chunk
#define NCHUNK (D_IN / KC)          // 16

__device__ __forceinline__ void issue_chunk(const float* __restrict__ x,
                                            int rowBlk, int kc,
                                            uint32_t ldsBase, int tid) {
    // 128 rows x 32 floats = 1024 float4 slots; 4 async b128 per thread.
    #pragma unroll
    for (int i = 0; i < 4; ++i) {
        int idx = tid + i * 256;        // 0..1023
        int r   = idx >> 3;             // 8 float4 per row
        int c4  = idx & 7;
        int grow = rowBlk + r;
        if (grow > N_NODES - 1) grow = N_NODES - 1;   // clamp: uniform issue count
        const float* g = x + (long long)grow * D_IN + kc + c4 * 4;
        async_copy_b128(ldsBase + (uint32_t)idx * 16u, g);
    }
}

__global__ __launch_bounds__(256)
void gemm1_wmma(const float* __restrict__ x, const float* __restrict__ W1,
                const float* __restrict__ b1, float* __restrict__ h1) {
    __shared__ __align__(16) float lds[2][128 * KC];   // 2 x 16 KB
    const int tid  = threadIdx.x;
    const int lane = tid & 31;
    const int wave = tid >> 5;
    const int lm   = lane & 15;     // row/col within 16
    const int lh   = lane >> 4;     // half-wave select (K pair / M offset)
    const int rowBlk = blockIdx.x * 128;
    const int rowW   = rowBlk + wave * 16;
    const bool live  = (rowW < N_NODES);   // wave-uniform (N_NODES % 16 == 0)

    const uint32_t base0 = (uint32_t)(uintptr_t)&lds[0][0];
    const uint32_t base1 = (uint32_t)(uintptr_t)&lds[1][0];

    v8f acc[4] = {};

    issue_chunk(x, rowBlk, 0, base0, tid);

    for (int c = 0; c < NCHUNK; ++c) {
        if (c < NCHUNK - 1) {
            // prefetch next chunk into the other buffer
            issue_chunk(x, rowBlk, (c + 1) * KC, (c & 1) ? base0 : base1, tid);
            wait_async_le4();           // chunk c landed; c+1 still in flight
        } else {
            wait_async_0();
        }
        __syncthreads();

        if (live) {
            const float* arow = &lds[c & 1][(wave * 16 + lm) * KC];
            #pragma unroll
            for (int k = 0; k < KC; k += 4) {
                int kb = k + 2 * lh;                // this lane-half's K pair
                v2f a;
                a.x = arow[kb];
                a.y = arow[kb + 1];
                const float* Bk = W1 + (long long)(c * KC + kb) * D_HID;
                #pragma unroll
                for (int t = 0; t < 4; ++t) {       // 4 independent WMMA chains
                    v2f b;
                    b.x = Bk[t * 16 + lm];
                    b.y = Bk[D_HID + t * 16 + lm];
                    acc[t] = __builtin_amdgcn_wmma_f32_16x16x4_f32(
                        false, a, false, b, (short)0, acc[t], false, false);
                }
            }
        }
        __syncthreads();    // readers done before buffer is refilled
    }

    if (live) {
        #pragma unroll
        for (int t = 0; t < 4; ++t) {
            int col    = t * 16 + lm;
            float bias = b1[col];
            #pragma unroll
            for (int r = 0; r < 8; ++r) {
                int m = rowW + r + 8 * lh;
                h1[(long long)m * D_HID + col] = acc[t][r] + bias;
            }
        }
    }
}

// ---------------------------------------------------------------------------
// GEMM2: h2[N,40] = relu(agg1[N,64]) @ W2[64,40] + b2   (fp32 WMMA 16x16x4)
// 3 C tiles (48 cols), cols >= 40 masked on B-load and store.
// ---------------------------------------------------------------------------
__global__ __launch_bounds__(256)
void gemm2_wmma(const float* __restrict__ agg1, const float* __restrict__ W2,
                const float* __restrict__ b2, float* __restrict__ h2) {
    const int lane = threadIdx.x & 31;
    const int wave = threadIdx.x >> 5;
    const int lm   = lane & 15;
    const int lh   = lane >> 4;
    const int rowW = blockIdx.x * 128 + wave * 16;
    if (rowW >= N_NODES) return;       // wave-uniform

    v8f acc[3] = {};
    const float* arow = agg1 + (long long)(rowW + lm) * D_HID;

    #pragma unroll 4
    for (int k = 0; k < D_HID; k += 4) {
        int kb = k + 2 * lh;
        v2f a;
        a.x = fmaxf(arow[kb],     0.f);   // fused relu
        a.y = fmaxf(arow[kb + 1], 0.f);
        #pragma unroll
        for (int t = 0; t < 3; ++t) {
            int col = t * 16 + lm;
            v2f b;
            b.x = (col < D_OUT) ? W2[(long long)kb * D_OUT + col]       : 0.f;
            b.y = (col < D_OUT) ? W2[(long long)(kb + 1) * D_OUT + col] : 0.f;
            acc[t] = __builtin_amdgcn_wmma_f32_16x16x4_f32(
                false, a, false, b, (short)0, acc[t], false, false);
        }
    }

    #pragma unroll
    for (int t = 0; t < 3; ++t) {
        int col = t * 16 + lm;
        if (col < D_OUT) {
            float bias = b2[col];
            #pragma unroll
            for (int r = 0; r < 8; ++r) {
                int m = rowW + r + 8 * lh;
                h2[(long long)m * D_OUT + col] = acc[t][r] + bias;
            }
        }
    }
}

// ---------------------------------------------------------------------------
// Edge gather-scale-scatter-add: agg[dst,:] += w[e] * h[src,:]
// One lane per (edge, feature); features contiguous in a wave -> coalesced
// gather and coalesced L2-resident global_atomic_add_f32.
// ---------------------------------------------------------------------------
template <int D>
__global__ __launch_bounds__(256)
void scatter_edges(const float* __restrict__ h, const int* __restrict__ src,
                   const int* __restrict__ dst, const float* __restrict__ w,
                   float* __restrict__ agg) {
    int tid = blockIdx.x * 256 + threadIdx.x;
    if (tid >= N_EDGES * D) return;
    int e = tid / D;
    int f = tid % D;
    float v = w[e] * h[(long long)src[e] * D + f];
    (void)__hip_atomic_fetch_add(&agg[(long long)dst[e] * D + f], v,
                                 __ATOMIC_RELAXED, __HIP_MEMORY_SCOPE_AGENT);
}

// ---------------------------------------------------------------------------
// Row-wise log_softmax over 40 classes: one wave32 per node.
// Lane l covers cols l and l+32 (<40); wave32 shfl_xor reductions.
// ---------------------------------------------------------------------------
__global__ __launch_bounds__(256)
void logsoftmax_rows(const float* __restrict__ agg2, float* __restrict__ out) {
    const int lane = threadIdx.x & 31;
    const int wave = threadIdx.x >> 5;
    const int node = blockIdx.x * 8 + wave;
    if (node >= N_NODES) return;
    const float* row = agg2 + (long long)node * D_OUT;

    float v0 = row[lane];
    bool  has2 = (lane + 32) < D_OUT;
    float v1 = has2 ? row[lane + 32] : -3.402823466e+38f;

    float m = fmaxf(v0, v1);
    #pragma unroll
    for (int off = 16; off > 0; off >>= 1)
        m = fmaxf(m, __shfl_xor(m, off, 32));

    float s = __expf(v0 - m) + (has2 ? __expf(v1 - m) : 0.f);
    #pragma unroll
    for (int off = 16; off > 0; off >>= 1)
        s += __shfl_xor(s, off, 32);

    float lse = m + __logf(s);
    out[(long long)node * D_OUT + lane] = v0 - lse;
    if (has2)
        out[(long long)node * D_OUT + lane + 32] = v1 - lse;
}

// ---------------------------------------------------------------------------
extern "C" void kernel_launch(void* const* d_in, const int* in_sizes, int n_in,
                              void* d_out, int out_size, void* d_ws, size_t ws_size,
                              hipStream_t stream) {
    const float* x     = (const float*)d_in[0];
    const int*   edge  = (const int*)  d_in[1];   // [2, E]
    const float* w_mul = (const float*)d_in[2];
    const float* W1    = (const float*)d_in[3];
    const float* b1    = (const float*)d_in[4];
    const float* W2    = (const float*)d_in[5];
    const float* b2    = (const float*)d_in[6];
    const int* src = edge;
    const int* dst = edge + N_EDGES;

    // workspace layout (all fp32): h1 | agg1 | h2 | agg2  (~41.6 MB)
    float* h1   = (float*)d_ws;
    float* agg1 = h1   + (size_t)N_NODES * D_HID;
    float* h2   = agg1 + (size_t)N_NODES * D_HID;
    float* agg2 = h2   + (size_t)N_NODES * D_OUT;

    hipMemsetAsync(agg1, 0, (size_t)N_NODES * D_HID * sizeof(float), stream);
    hipMemsetAsync(agg2, 0, (size_t)N_NODES * D_OUT * sizeof(float), stream);

    const int gemmBlocks = (N_NODES + 127) / 128;   // 391

    gemm1_wmma<<<gemmBlocks, 256, 0, stream>>>(x, W1, b1, h1);

    scatter_edges<D_HID><<<(N_EDGES * D_HID + 255) / 256, 256, 0, stream>>>(
        h1, src, dst, w_mul, agg1);

    gemm2_wmma<<<gemmBlocks, 256, 0, stream>>>(agg1, W2, b2, h2);

    scatter_edges<D_OUT><<<(N_EDGES * D_OUT + 255) / 256, 256, 0, stream>>>(
        h2, src, dst, w_mul, agg2);

    logsoftmax_rows<<<(N_NODES + 7) / 8, 256, 0, stream>>>(agg2, (float*)d_out);
}